// Spatial_Grounding_Indicator_43370579755476
// MI455X (gfx1250) — compile-verified
//
#include <hip/hip_runtime.h>
#include <math.h>

#define B_      32
#define L_      2048
#define D_      1024
#define FRAMES  64
#define OFRAME  32
#define OTOPK   5

typedef __attribute__((ext_vector_type(2))) float v2f;
typedef __attribute__((ext_vector_type(8))) float v8f;

// ---------------------------------------------------------------------------
// Phase 1: att[b,l] = dot(q[b,:], V[b,l,:])  via V_WMMA_F32_16X16X4_F32.
// One wave (32 lanes) owns a 16-row tile of V.  Per WMMA step over K=4:
//   A[m,k]: lane (m | (khalf?16:0)) holds V[row m, d+kh], V[row m, d+kh+1]
//   B[k,n]: q[d+k] broadcast to every column n (mirrored per-lane K map)
// All 16 columns of the f32 16x16 accumulator equal the row dot-products;
// lane 0 holds rows 0..7 (vgpr r = row r, N=0), lane 16 holds rows 8..15.
// v_local streams once (256 MB > 192 MB L2) -> non-temporal loads; q stays
// cached.  Two accumulators break the WMMA RAW chain.
// ---------------------------------------------------------------------------
__global__ __launch_bounds__(128)
void sgi_gemv_wmma(const float* __restrict__ q,
                   const float* __restrict__ v,
                   float* __restrict__ att) {
  const int lane = threadIdx.x & 31;
  const int wave = threadIdx.x >> 5;
  const int tile = blockIdx.x * 4 + wave;     // 16-row tile id
  const int tiles_per_b = L_ / 16;            // 128
  const int b  = tile / tiles_per_b;
  const int l0 = (tile % tiles_per_b) * 16;
  const int m  = lane & 15;
  const int kh = (lane >> 4) << 1;            // 0 or 2

  // SGPR base per batch + 32-bit per-lane element offset (256MB fits 32-bit)
  const float* __restrict__ vb = v + (size_t)b * L_ * D_;
  const int voff = (l0 + m) * D_ + kh;
  const float* __restrict__ qb = q + b * D_;

  v8f acc0 = {};
  v8f acc1 = {};
  #pragma unroll 4
  for (int d = 0; d < D_; d += 8) {
    v2f a0 = __builtin_nontemporal_load((const v2f*)(vb + voff + d));
    v2f a1 = __builtin_nontemporal_load((const v2f*)(vb + voff + d + 4));
    v2f b0 = { qb[d + kh],     qb[d + kh + 1] };
    v2f b1 = { qb[d + 4 + kh], qb[d + 4 + kh + 1] };
    acc0 = __builtin_amdgcn_wmma_f32_16x16x4_f32(
        false, a0, false, b0, (short)0, acc0, false, false);
    acc1 = __builtin_amdgcn_wmma_f32_16x16x4_f32(
        false, a1, false, b1, (short)0, acc1, false, false);
  }

  v8f fin;
  #pragma unroll
  for (int r = 0; r < 8; ++r) fin[r] = acc0[r] + acc1[r];

  float* dst = att + (size_t)b * L_ + l0;     // 64B-aligned (l0 % 16 == 0)
  if (lane == 0) {
    *(v8f*)dst = fin;                          // rows 0..7
  } else if (lane == 16) {
    *(v8f*)(dst + 8) = fin;                    // rows 8..15
  }
}

// ---------------------------------------------------------------------------
// Phase 2: one block per batch b.  Softmax over L, mask renorm, Gumbel split
// (y == y_hard exactly since y_soft - stop_grad(y_soft) == 0), per-frame
// stable top-5 / bottom-5 (matches JAX stable argsort tie-breaks), box gather.
// ---------------------------------------------------------------------------
__global__ __launch_bounds__(256)
void sgi_finalize(const float* __restrict__ att,
                  const float* __restrict__ v_box,
                  const float* __restrict__ attn_mask,
                  const float* __restrict__ gumbel,
                  float* __restrict__ out_box,
                  float* __restrict__ pos_mask_out,
                  float* __restrict__ neg_mask_out) {
  __shared__ float s_buf[L_];                 // exp -> masked prob (in place)
  __shared__ float s_pos[L_];
  __shared__ float s_neg[L_];
  __shared__ float red[256];
  __shared__ int   s_posidx[FRAMES * OTOPK];

  const int b = blockIdx.x;
  const int t = threadIdx.x;
  const size_t base = (size_t)b * L_;

  // ---- pass A: max over att[b,:] ----
  float lmax = -INFINITY;
  for (int l = t; l < L_; l += 256) {
    float a = att[base + l];
    s_buf[l] = a;
    lmax = fmaxf(lmax, a);
  }
  red[t] = lmax;
  __syncthreads();
  for (int s = 128; s > 0; s >>= 1) {
    if (t < s) red[t] = fmaxf(red[t], red[t + s]);
    __syncthreads();
  }
  const float gmax = red[0];
  __syncthreads();

  // ---- pass B: exp and sum ----
  float lsum = 0.f;
  for (int l = t; l < L_; l += 256) {
    float e = expf(s_buf[l] - gmax);
    s_buf[l] = e;
    lsum += e;
  }
  red[t] = lsum;
  __syncthreads();
  for (int s = 128; s > 0; s >>= 1) {
    if (t < s) red[t] += red[t + s];
    __syncthreads();
  }
  const float sum1 = red[0];
  __syncthreads();

  // ---- pass C: p = softmax * attn_mask, masked sum ----
  float lsum2 = 0.f;
  for (int l = t; l < L_; l += 256) {
    float p = (s_buf[l] / sum1) * attn_mask[base + l];
    s_buf[l] = p;
    lsum2 += p;
  }
  red[t] = lsum2;
  __syncthreads();
  for (int s = 128; s > 0; s >>= 1) {
    if (t < s) red[t] += red[t + s];
    __syncthreads();
  }
  const float sum2 = red[0] + 1e-12f;
  __syncthreads();

  // ---- pass D: score, Gumbel hard split, masked pos/neg scores ----
  for (int l = t; l < L_; l += 256) {
    float am = attn_mask[base + l];
    float score = (am == 0.f) ? -10000.0f : (s_buf[l] / sum2);
    float g0 = gumbel[((size_t)b * 2 + 0) * L_ + l];
    float g1 = gumbel[((size_t)b * 2 + 1) * L_ + l];
    bool pos = (score + g0) >= (score + g1);   // argmax(axis=1), ties -> 0
    float pmsk = pos ? am : 0.f;
    float nmsk = pos ? 0.f : am;
    s_pos[l] = (pmsk == 0.f) ? -10000.0f : score * pmsk;
    s_neg[l] = (nmsk == 0.f) ? -10000.0f : score * nmsk;
  }
  __syncthreads();

  // ---- per-frame stable top-5 (pos, desc) / bottom-5 (neg, asc) ----
  if (t < 2 * FRAMES) {
    const bool is_pos = (t < FRAMES);
    const int  f = is_pos ? t : (t - FRAMES);
    const float* sc = (is_pos ? s_pos : s_neg) + f * OFRAME;
    float* mout = (is_pos ? pos_mask_out : neg_mask_out) + base + f * OFRAME;

    unsigned used = 0u;
    int idxs[OTOPK];
    #pragma unroll
    for (int p = 0; p < OTOPK; ++p) {
      float best = is_pos ? -INFINITY : INFINITY;
      int bi = 0;
      for (int j = 0; j < OFRAME; ++j) {
        if (used & (1u << j)) continue;
        float vv = sc[j];
        bool take = is_pos ? (vv > best) : (vv < best);
        if (take) { best = vv; bi = j; }
      }
      used |= (1u << bi);
      idxs[p] = bi;
    }
    for (int j = 0; j < OFRAME; ++j) {
      float mv = ((used >> j) & 1u) ? 1.f : 0.f;
      if (!is_pos) mv *= attn_mask[base + f * OFRAME + j];
      mout[j] = mv;
    }
    if (is_pos) {
      #pragma unroll
      for (int i = 1; i < OTOPK; ++i) {        // insertion sort, ascending
        int key = idxs[i], j = i - 1;
        while (j >= 0 && idxs[j] > key) { idxs[j + 1] = idxs[j]; --j; }
        idxs[j + 1] = key;
      }
      #pragma unroll
      for (int p = 0; p < OTOPK; ++p) s_posidx[f * OTOPK + p] = idxs[p];
    }
  }
  __syncthreads();

  // ---- gather boxes: (B, FRAMES*OTOPK, 4) ----
  for (int i = t; i < FRAMES * OTOPK; i += 256) {
    int f = i / OTOPK;
    int gidx = s_posidx[i] + f * OFRAME;
    const float* src = v_box + (base + gidx) * 4;
    float* dst = out_box + ((size_t)b * FRAMES * OTOPK + i) * 4;
    dst[0] = src[0]; dst[1] = src[1]; dst[2] = src[2]; dst[3] = src[3];
  }
}

extern "C" void kernel_launch(void* const* d_in, const int* in_sizes, int n_in,
                              void* d_out, int out_size, void* d_ws, size_t ws_size,
                              hipStream_t stream) {
  const float* q         = (const float*)d_in[0];   // (B,1,D)
  const float* v_local   = (const float*)d_in[1];   // (B,L,D)
  const float* v_box     = (const float*)d_in[2];   // (B,L,4)
  // d_in[3] v_mask: unused by the reference
  const float* attn_mask = (const float*)d_in[4];   // (B,L)
  const float* gumbel    = (const float*)d_in[5];   // (B,2,L)
  // d_in[6..9]: f_topk (unused), o_topk=5, frame_num=64, o_frame_num=32 (fixed)

  float* att = (float*)d_ws;                         // B*L floats = 256 KB

  float* out      = (float*)d_out;
  float* out_box  = out;                                         // 32*320*4
  float* pos_out  = out + (size_t)B_ * FRAMES * OTOPK * 4;       // 32*2048
  float* neg_out  = pos_out + (size_t)B_ * L_;                   // 32*2048

  const int tiles = (B_ * L_) / 16;                  // 4096 waves
  sgi_gemv_wmma<<<tiles / 4, 128, 0, stream>>>(q, v_local, att);
  sgi_finalize<<<B_, 256, 0, stream>>>(att, v_box, attn_mask, gumbel,
                                       out_box, pos_out, neg_out);
}